// RefLocal4_65670049956458
// MI455X (gfx1250) — compile-verified
//
#include <hip/hip_runtime.h>
#include <hip/hip_bf16.h>

// Problem constants (match reference setup_inputs)
#define B_    4
#define H_    128
#define W_    128
#define C_    64     // channels == BIN == 64
#define TILE  16     // pixels per workgroup along W
#define PAD   2      // K_SIZE/2
#define KW    20     // TILE + 2*PAD key/value columns held in LDS
#define NDY   5
#define NSLOT 26     // 25 window slots + self
#define LPAD  32     // padded L row stride; column 31 = trash sink
#define TRASH 31

typedef float v2f __attribute__((ext_vector_type(2)));
typedef float v8f __attribute__((ext_vector_type(8)));

// Full-precision f32 WMMA: D(16x16,f32) = A(16x4,f32) x B(4x16,f32) + C
__device__ __forceinline__ v8f wmma_k4(v2f a, v2f b, v8f c) {
    return __builtin_amdgcn_wmma_f32_16x16x4_f32(
        /*neg_a=*/false, a, /*neg_b=*/false, b,
        /*c_mod=*/(short)0, c, /*reuse_a=*/false, /*reuse_b=*/false);
}

// Banded probability matrix entry A_t[m][j] for the aggregation stage.
// t=0 tile covers key w = w0-2+j ; t=1 tile covers key w = w0+6+j.
// Each (pixel m, dx) lives in exactly one tile (t=0 iff m+dx+2 <= 15).
// Called with compile-time t (unrolled loop) -> branchless cndmask selects.
__device__ __forceinline__ float band_p(const float (*L)[LPAD], int m, int j,
                                        int dy, int t) {
    int d = j - m;
    if (t == 0) {
        // dx = d - 2 in [-2,2]  ->  slot dy*5 + d
        return (d >= 0 && d <= 4) ? L[m][dy * 5 + d] : 0.0f;
    } else {
        // dx = d + 6 in [-2,2] and overflow case (j >= 8) -> slot dy*5 + d + 8
        return (d >= -8 && d <= -4 && j >= 8) ? L[m][dy * 5 + d + 8] : 0.0f;
    }
}

__global__ __launch_bounds__(128, 1)
void local_attn_wmma(const float* __restrict__ main_in,
                     const float* __restrict__ mval_in,
                     const float* __restrict__ ref_in,
                     const float* __restrict__ rval_in,
                     float* __restrict__ out) {
    __shared__ __align__(16) float Qs [TILE][C_];        // main tile (queries)
    __shared__ __align__(16) float MVs[TILE][C_];        // main_value tile
    __shared__ __align__(16) float Ks [NDY][KW][C_];     // ref rows, zero-padded
    __shared__ __align__(16) float Vs [NDY][KW][C_];     // ref_value rows, zero-padded
    __shared__ float L[TILE][LPAD];                      // logits -> probabilities

    const int wg = blockIdx.x;            // B_*H_*(W_/TILE) blocks
    const int wt = wg & 7;                // w-tile index (W_/TILE == 8)
    const int h  = (wg >> 3) & (H_ - 1);
    const int b  = wg >> 10;
    const int w0 = wt * TILE;

    const int tid  = threadIdx.x;         // 0..127
    const int lane = tid & 31;            // wave32 lane
    const int wave = tid >> 5;            // 0..3
    const int lo   = lane & 15;           // row/col within 16
    const int hi   = lane >> 4;           // 0/1 half selector

    // ---------------- cooperative loads ----------------
    {
        const size_t base = (((size_t)b * H_ + h) * W_ + w0) * C_;   // contiguous 1024 f32
        const float4* gq = reinterpret_cast<const float4*>(main_in + base);
        const float4* gm = reinterpret_cast<const float4*>(mval_in + base);
        float4* sq = reinterpret_cast<float4*>(&Qs[0][0]);
        float4* sm = reinterpret_cast<float4*>(&MVs[0][0]);
        sq[tid]       = gq[tid];
        sq[tid + 128] = gq[tid + 128];
        sm[tid]       = gm[tid];
        sm[tid + 128] = gm[tid + 128];
    }
    // 5 rows x 20 cols of ref / ref_value, zero-padded at image borders.
    // 4 threads cooperate per (dy,col) position; 16 f32 (4 float4) each.
    for (int p = (tid >> 2); p < NDY * KW; p += 32) {
        const int dy  = p / KW;
        const int col = p % KW;
        const int gh  = h + dy - PAD;
        const int gw  = w0 + col - PAD;
        const bool ok = (gh >= 0) && (gh < H_) && (gw >= 0) && (gw < W_);
        const size_t gbase = (((size_t)b * H_ + (ok ? gh : 0)) * W_ + (ok ? gw : 0)) * C_;
        const float4* srcK = reinterpret_cast<const float4*>(ref_in  + gbase);
        const float4* srcV = reinterpret_cast<const float4*>(rval_in + gbase);
        float4* dK = reinterpret_cast<float4*>(&Ks[dy][col][0]);
        float4* dV = reinterpret_cast<float4*>(&Vs[dy][col][0]);
        const int sub = tid & 3;
        const float4 z = make_float4(0.f, 0.f, 0.f, 0.f);
        #pragma unroll
        for (int j = 0; j < 4; ++j) {
            const int e = sub * 4 + j;
            dK[e] = ok ? srcK[e] : z;
            dV[e] = ok ? srcV[e] : z;
        }
    }
    __syncthreads();

    // ---------------- stage 1: QK^T logits via f32 WMMA ----------------
    // job = dy + 5*t ; tile t=0: key cols w0-2..w0+13 ; t=1: w0+6..w0+21.
    for (int job = wave; job < 2 * NDY; job += 4) {
        const int dy = job % NDY;
        const int t  = job / NDY;
        // A row = pixel (lo); B col = key position (lo), clamped: clamped cols
        // (t=1, j>=12 -> w > w0+17) are never consumed by any pixel's band.
        const int kc = min(t * 8 + lo, KW - 1);
        const float* Arow = &Qs[lo][0];
        const float* Brow = &Ks[dy][kc][0];
        v8f acc = {0.f, 0.f, 0.f, 0.f, 0.f, 0.f, 0.f, 0.f};
        #pragma unroll
        for (int kk = 0; kk < C_; kk += 4) {
            // A 16x4 f32 layout: lane m, VGPR0/1 = K = kk+2*hi, kk+2*hi+1
            v2f a = *reinterpret_cast<const v2f*>(Arow + kk + 2 * hi);
            // B 4x16 f32 layout: lane n, VGPR0/1 = K = kk+2*hi, kk+2*hi+1
            v2f bb = *reinterpret_cast<const v2f*>(Brow + kk + 2 * hi);
            acc = wmma_k4(a, bb, acc);
        }
        // C layout: acc[r] = S[m = r + 8*hi][n = lo].
        // Branchless scatter: invalid (lane,r) combos write to trash column 31
        // so there is no exec-mask divergence around the LDS stores.
        #pragma unroll
        for (int r = 0; r < 8; ++r) {
            const int mm = r + 8 * hi;
            const int d  = lo - mm;
            const int slot0 = dy * 5 + d;                       // t=0: dx = d-2
            const bool v0 = (t == 0) & (d >= 0) & (d <= 4);
            const bool v1 = (t != 0) & (d >= -8) & (d <= -4) & (lo >= 8);
            const int slot = v0 ? slot0 : (v1 ? slot0 + 8 : TRASH);
            L[mm][slot] = acc[r];
        }
    }
    // self logit: dot(main, main)
    if (tid < TILE) {
        float s = 0.f;
        #pragma unroll 8
        for (int c = 0; c < C_; ++c) s += Qs[tid][c] * Qs[tid][c];
        L[tid][NSLOT - 1] = s;
    }
    __syncthreads();

    // ---------------- stage 2: softmax over 26 slots ----------------
    if (tid < TILE) {
        float mx = L[tid][0];
        #pragma unroll
        for (int k = 1; k < NSLOT; ++k) mx = fmaxf(mx, L[tid][k]);
        float p[NSLOT];
        float sum = 0.f;
        #pragma unroll
        for (int k = 0; k < NSLOT; ++k) {
            p[k] = expf(L[tid][k] - mx);
            sum += p[k];
        }
        const float inv = 1.0f / sum;
        #pragma unroll
        for (int k = 0; k < NSLOT; ++k) L[tid][k] = p[k] * inv;
    }
    __syncthreads();

    // ---------------- stage 3: banded P x V aggregation via f32 WMMA ----------------
    const int c0 = wave * 16;     // each wave owns one 16-channel output tile
    v8f acc = {0.f, 0.f, 0.f, 0.f, 0.f, 0.f, 0.f, 0.f};
    for (int dy = 0; dy < NDY; ++dy) {
        #pragma unroll
        for (int t = 0; t < 2; ++t) {
            #pragma unroll
            for (int kk = 0; kk < 16; kk += 4) {
                const int k0 = kk + 2 * hi;
                v2f a;
                a.x = band_p(L, lo, k0,     dy, t);
                a.y = band_p(L, lo, k0 + 1, dy, t);
                // B[k][n] = Vs[dy][t*8 + k][c0 + n] (clamped cols have zero A weight)
                const int col0 = min(t * 8 + k0,     KW - 1);
                const int col1 = min(t * 8 + k0 + 1, KW - 1);
                v2f bb;
                bb.x = Vs[dy][col0][c0 + lo];
                bb.y = Vs[dy][col1][c0 + lo];
                acc = wmma_k4(a, bb, acc);
            }
        }
    }
    // self term + store: acc[r] = out[m = r + 8*hi][n = lo] for channel tile c0
    #pragma unroll
    for (int r = 0; r < 8; ++r) {
        const int mm = r + 8 * hi;
        const float val = acc[r] + L[mm][NSLOT - 1] * MVs[mm][c0 + lo];
        out[(((size_t)b * H_ + h) * W_ + (w0 + mm)) * C_ + c0 + lo] = val;
    }
}

extern "C" void kernel_launch(void* const* d_in, const int* in_sizes, int n_in,
                              void* d_out, int out_size, void* d_ws, size_t ws_size,
                              hipStream_t stream) {
    (void)in_sizes; (void)n_in; (void)d_ws; (void)ws_size; (void)out_size;
    const float* main_in = (const float*)d_in[0];
    const float* mval_in = (const float*)d_in[1];
    const float* ref_in  = (const float*)d_in[2];
    const float* rval_in = (const float*)d_in[3];
    float* out = (float*)d_out;

    const int grid = B_ * H_ * (W_ / TILE);   // 4096 workgroups
    local_attn_wmma<<<grid, 128, 0, stream>>>(main_in, mval_in, ref_in, rval_in, out);
}